// SPDBatchNorm_49933289783348
// MI455X (gfx1250) — compile-verified
//
#include <hip/hip_runtime.h>
#include <math.h>

#define NDIM 64
#define LDA  65                 // LDS row stride (padded, bank-conflict free)
#define BUF  (NDIM * LDA)       // 4160 floats per LDS matrix
#define BATCH 8192
#define NSWEEPS 8

typedef __attribute__((ext_vector_type(2))) float v2f;
typedef __attribute__((ext_vector_type(8))) float v8f;

// workspace layout (float offsets)
#define WS_G     0
#define WS_GS    4096
#define WS_GIS   8192
#define WS_TACC  12288
#define WS_M     16384
#define WS_VAR   20480
#define WS_ALPHA 20481
#define WS_W     20992          // BATCH*64 eigenvalues

// ---------------- helpers ----------------

__device__ __forceinline__ void pairpq(int r, int k, int* p, int* q) {
  int a, b;
  if (k == 0) { a = 63; b = r; }
  else { a = (r + k) % 63; b = (r - k + 63) % 63; }
  *p = a < b ? a : b;
  *q = a < b ? b : a;
}

__device__ __forceinline__ void ld_tile(float* dst, const float* __restrict__ src) {
  for (int i = threadIdx.x; i < NDIM * NDIM; i += blockDim.x)
    dst[(i >> 6) * LDA + (i & 63)] = src[i];
  __syncthreads();
}

__device__ __forceinline__ void st_tile(float* __restrict__ dstg, const float* src) {
  for (int i = threadIdx.x; i < NDIM * NDIM; i += blockDim.x)
    dstg[i] = src[(i >> 6) * LDA + (i & 63)];
  __syncthreads();
}

// C(64x64) = op(A) * op(B), fp32 via V_WMMA_F32_16X16X4_F32.
// 256-thread block = 8 waves; 16 output tiles -> 2 tiles/wave; K loop of 16 WMMAs.
__device__ void wave_gemm(const float* A, const float* B, float* C, bool tA, bool tB) {
  const int tid  = threadIdx.x;
  const int wave = tid >> 5;
  const int lane = tid & 31;
  const int half = lane >> 4;   // lanes 16..31 hold K+2 / M+8 per ISA layout
  const int l16  = lane & 15;
  for (int t = wave; t < 16; t += 8) {
    const int tm = (t >> 2) << 4;
    const int tn = (t & 3) << 4;
    v8f acc = {0.f, 0.f, 0.f, 0.f, 0.f, 0.f, 0.f, 0.f};
    for (int k0 = 0; k0 < NDIM; k0 += 4) {
      const int ka = k0 + half * 2;
      v2f a, b;
      if (!tA) {
        a.x = A[(tm + l16) * LDA + ka];
        a.y = A[(tm + l16) * LDA + ka + 1];
      } else {
        a.x = A[ka * LDA + tm + l16];
        a.y = A[(ka + 1) * LDA + tm + l16];
      }
      if (!tB) {
        b.x = B[ka * LDA + tn + l16];
        b.y = B[(ka + 1) * LDA + tn + l16];
      } else {
        b.x = B[(tn + l16) * LDA + ka];
        b.y = B[(tn + l16) * LDA + ka + 1];
      }
      acc = __builtin_amdgcn_wmma_f32_16x16x4_f32(false, a, false, b,
                                                  (short)0, acc, false, false);
    }
#pragma unroll
    for (int v = 0; v < 8; ++v)
      C[(tm + v + half * 8) * LDA + tn + l16] = acc[v];
  }
  __syncthreads();
}

// Parallel cyclic Jacobi eigensolver for symmetric 64x64 in LDS.
// On exit: diag(A) = eigenvalues, V columns = eigenvectors.
__device__ void jacobi_eigh(float* A, float* V) {
  const int tid = threadIdx.x;
  __shared__ float cs[32], sn[32];
  for (int i = tid; i < NDIM * NDIM; i += blockDim.x) {
    int r = i >> 6, c = i & 63;
    V[r * LDA + c] = (r == c) ? 1.0f : 0.0f;
  }
  __syncthreads();
  for (int sweep = 0; sweep < NSWEEPS; ++sweep) {
    for (int r = 0; r < 63; ++r) {
      if (tid < 32) {
        int p, q; pairpq(r, tid, &p, &q);
        float app = A[p * LDA + p], aqq = A[q * LDA + q], apq = A[p * LDA + q];
        float c = 1.0f, s = 0.0f;
        if (fabsf(apq) > 1e-12f * (fabsf(app) + fabsf(aqq)) + 1e-30f) {
          float tau = (aqq - app) / (2.0f * apq);
          float t = copysignf(1.0f, tau) / (fabsf(tau) + sqrtf(1.0f + tau * tau));
          c = rsqrtf(1.0f + t * t);
          s = t * c;
        }
        cs[tid] = c; sn[tid] = s;
      }
      __syncthreads();
      // column update: A <- A*J, V <- V*J  (32 pairs x 64 rows)
      for (int it = tid; it < 32 * NDIM; it += blockDim.x) {
        int k = it >> 6, row = it & 63;
        int p, q; pairpq(r, k, &p, &q);
        float c = cs[k], s = sn[k];
        float ap = A[row * LDA + p], aq = A[row * LDA + q];
        A[row * LDA + p] = c * ap - s * aq;
        A[row * LDA + q] = s * ap + c * aq;
        float vp = V[row * LDA + p], vq = V[row * LDA + q];
        V[row * LDA + p] = c * vp - s * vq;
        V[row * LDA + q] = s * vp + c * vq;
      }
      __syncthreads();
      // row update: A <- J^T*A
      for (int it = tid; it < 32 * NDIM; it += blockDim.x) {
        int k = it >> 6, col = it & 63;
        int p, q; pairpq(r, k, &p, &q);
        float c = cs[k], s = sn[k];
        float ap = A[p * LDA + col], aq = A[q * LDA + col];
        A[p * LDA + col] = c * ap - s * aq;
        A[q * LDA + col] = s * ap + c * aq;
      }
      __syncthreads();
    }
  }
}

#define F_SQRT  0
#define F_RSQRT 1
#define F_LOG   2
#define F_EXP   3
#define F_POW   4

// Out = V * diag(f(diag(Adiag)))
__device__ void col_scale_diag(const float* V, const float* Adiag, float* Out,
                               int mode, float p) {
  const int tid = threadIdx.x;
  __shared__ float fw[NDIM];
  if (tid < NDIM) {
    float w = Adiag[tid * LDA + tid];
    if (mode != F_EXP) w = fmaxf(w, 1e-20f);
    float r;
    switch (mode) {
      case F_SQRT:  r = sqrtf(w); break;
      case F_RSQRT: r = rsqrtf(w); break;
      case F_LOG:   r = logf(w); break;
      case F_EXP:   r = expf(w); break;
      default:      r = powf(w, p); break;
    }
    fw[tid] = r;
  }
  __syncthreads();
  for (int i = tid; i < NDIM * NDIM; i += blockDim.x) {
    int r = i >> 6, c = i & 63;
    Out[r * LDA + c] = V[r * LDA + c] * fw[c];
  }
  __syncthreads();
}

// ---------------- kernels ----------------

__global__ void k_init(float* __restrict__ ws) {
  const int tid = threadIdx.x;
  for (int i = tid; i < NDIM * NDIM; i += blockDim.x) {
    ws[WS_G + i] = 0.0f;
    ws[WS_TACC + i] = 0.0f;
  }
  if (tid == 0) { ws[WS_VAR] = 0.0f; ws[WS_ALPHA] = 0.0f; }
}

__global__ void __launch_bounds__(256) k_mean(const float* __restrict__ X,
                                              float* __restrict__ G) {
  __shared__ float acc[NDIM * NDIM];
  const int tid = threadIdx.x;
  for (int i = tid; i < NDIM * NDIM; i += blockDim.x) acc[i] = 0.0f;
  __syncthreads();
  const int per = BATCH / gridDim.x;
  const int b0 = blockIdx.x * per;
  for (int b = b0; b < b0 + per; ++b) {
    const float* src = X + (size_t)b * (NDIM * NDIM);
    for (int i = tid; i < NDIM * NDIM; i += blockDim.x) acc[i] += src[i];
  }
  __syncthreads();
  for (int i = tid; i < NDIM * NDIM; i += blockDim.x)
    atomicAdd(&G[i], acc[i] * (1.0f / BATCH));
}

__global__ void __launch_bounds__(256) k_sqrt_invsqrt(const float* __restrict__ Min,
                                                      float* __restrict__ Sout,
                                                      float* __restrict__ ISout) {
  extern __shared__ float smem[];
  float* A  = smem;
  float* V  = A + BUF;
  float* W1 = V + BUF;
  float* C  = W1 + BUF;
  ld_tile(A, Min);
  jacobi_eigh(A, V);
  col_scale_diag(V, A, W1, F_SQRT, 0.f);
  wave_gemm(W1, V, C, false, true);
  st_tile(Sout, C);
  col_scale_diag(V, A, W1, F_RSQRT, 0.f);
  wave_gemm(W1, V, C, false, true);
  st_tile(ISout, C);
}

__global__ void __launch_bounds__(256) k_karcher_log(const float* __restrict__ X,
                                                     const float* __restrict__ Gis,
                                                     float* __restrict__ Tacc) {
  extern __shared__ float smem[];
  float* Xs = smem;
  float* Gi = Xs + BUF;
  float* T1 = Gi + BUF;
  float* V  = T1 + BUF;
  const int b = blockIdx.x;
  ld_tile(Xs, X + (size_t)b * (NDIM * NDIM));
  ld_tile(Gi, Gis);
  wave_gemm(Gi, Xs, T1, false, false);   // Gis @ X
  wave_gemm(T1, Gi, Xs, false, false);   // S = Gis X Gis
  jacobi_eigh(Xs, V);
  col_scale_diag(V, Xs, T1, F_LOG, 0.f);
  wave_gemm(T1, V, Gi, false, true);     // logm(S)
  for (int i = threadIdx.x; i < NDIM * NDIM; i += blockDim.x)
    atomicAdd(&Tacc[i], Gi[(i >> 6) * LDA + (i & 63)] * (1.0f / BATCH));
}

__global__ void __launch_bounds__(256) k_middle(const float* __restrict__ Tmean,
                                                const float* __restrict__ Gs,
                                                const float* __restrict__ Rmean,
                                                const float* __restrict__ gamma,
                                                float* __restrict__ Mout) {
  extern __shared__ float smem[];
  float* A  = smem;
  float* V  = A + BUF;
  float* W1 = V + BUF;
  float* W2 = W1 + BUF;
  float* W3 = W2 + BUF;
  // E = expm(T)
  ld_tile(A, Tmean);
  jacobi_eigh(A, V);
  col_scale_diag(V, A, W1, F_EXP, 0.f);
  wave_gemm(W1, V, W2, false, true);       // E
  // Bk = Gs E Gs
  ld_tile(W1, Gs);
  wave_gemm(W1, W2, W3, false, false);
  wave_gemm(W3, W1, W2, false, false);     // Bk in W2
  // As / Ais from running mean
  ld_tile(A, Rmean);
  jacobi_eigh(A, V);
  col_scale_diag(V, A, W1, F_SQRT, 0.f);
  wave_gemm(W1, V, W3, false, true);       // As in W3
  col_scale_diag(V, A, W1, F_RSQRT, 0.f);
  wave_gemm(W1, V, A, false, true);        // Ais in A
  // P = Ais Bk Ais
  wave_gemm(A, W2, W1, false, false);
  wave_gemm(W1, A, W2, false, false);      // P in W2
  // Pg = P^gamma
  float g = gamma[0];
  jacobi_eigh(W2, V);
  col_scale_diag(V, W2, W1, F_POW, g);
  wave_gemm(W1, V, A, false, true);        // Pg in A
  // rm = As Pg As
  wave_gemm(W3, A, W1, false, false);
  wave_gemm(W1, W3, W2, false, false);     // rm in W2
  // M = invsqrtm(rm)
  jacobi_eigh(W2, V);
  col_scale_diag(V, W2, W1, F_RSQRT, 0.f);
  wave_gemm(W1, V, A, false, true);
  st_tile(Mout, A);
}

__global__ void __launch_bounds__(256) k_whiten_eigh(const float* __restrict__ X,
                                                     const float* __restrict__ M,
                                                     float* __restrict__ Vout,
                                                     float* __restrict__ wout,
                                                     float* __restrict__ var_acc) {
  extern __shared__ float smem[];
  float* Xs = smem;
  float* Ms = Xs + BUF;
  float* T1 = Ms + BUF;
  float* V  = T1 + BUF;
  __shared__ float red[NDIM];
  const int b = blockIdx.x;
  const int tid = threadIdx.x;
  ld_tile(Xs, X + (size_t)b * (NDIM * NDIM));
  ld_tile(Ms, M);
  wave_gemm(Ms, Xs, T1, false, false);
  wave_gemm(T1, Ms, Xs, false, false);     // W = M X M
  jacobi_eigh(Xs, V);
  st_tile(Vout + (size_t)b * (NDIM * NDIM), V);
  if (tid < NDIM) {
    float w = fmaxf(Xs[tid * LDA + tid], 1e-20f);
    wout[(size_t)b * NDIM + tid] = w;
    float l = logf(w);
    red[tid] = l * l;
  }
  __syncthreads();
  if (tid == 0) {
    float s = 0.f;
    for (int i = 0; i < NDIM; ++i) s += red[i];
    atomicAdd(var_acc, s);
  }
}

__global__ void k_scalars(const float* __restrict__ var_acc,
                          const float* __restrict__ running_var,
                          const float* __restrict__ raw_std,
                          const float* __restrict__ gamma,
                          float* __restrict__ alpha_out) {
  float var = var_acc[0] * (1.0f / BATCH);
  float g = gamma[0];
  float rv = (1.0f - g) * running_var[0] + g * var;
  float x = raw_std[0];
  float sp = (x > 20.0f) ? x : log1pf(expf(x));   // softplus
  float std = sp + 1e-3f;
  alpha_out[0] = sqrtf(std / (rv + 1e-5f));
}

__global__ void __launch_bounds__(256) k_reconstruct(const float* __restrict__ Vg,
                                                     const float* __restrict__ wg,
                                                     const float* __restrict__ alpha,
                                                     const float* __restrict__ Q,
                                                     float* __restrict__ Out) {
  extern __shared__ float smem[];
  float* V  = smem;
  float* T1 = V + BUF;
  float* Qs = T1 + BUF;
  float* Y  = Qs + BUF;
  __shared__ float fw[NDIM];
  const int b = blockIdx.x;
  const int tid = threadIdx.x;
  ld_tile(V, Vg + (size_t)b * (NDIM * NDIM));
  ld_tile(Qs, Q);
  if (tid < NDIM) fw[tid] = powf(wg[(size_t)b * NDIM + tid], alpha[0]);
  __syncthreads();
  for (int i = tid; i < NDIM * NDIM; i += blockDim.x) {
    int r = i >> 6, c = i & 63;
    T1[r * LDA + c] = V[r * LDA + c] * fw[c];
  }
  __syncthreads();
  wave_gemm(T1, V, Y, false, true);     // Xn = V diag(w^a) V^T
  wave_gemm(Qs, Y, T1, true, false);    // Q^T Xn
  wave_gemm(T1, Qs, V, false, false);   // (Q^T Xn) Q
  st_tile(Out + (size_t)b * (NDIM * NDIM), V);
}

// ---------------- host launcher ----------------

extern "C" void kernel_launch(void* const* d_in, const int* in_sizes, int n_in,
                              void* d_out, int out_size, void* d_ws, size_t ws_size,
                              hipStream_t stream) {
  (void)in_sizes; (void)n_in; (void)out_size; (void)ws_size;
  const float* X       = (const float*)d_in[0];   // [B,64,64]
  const float* raw_std = (const float*)d_in[1];   // scalar
  const float* Q       = (const float*)d_in[2];   // [64,64]
  const float* rmean   = (const float*)d_in[3];   // [64,64]
  const float* rvar    = (const float*)d_in[4];   // [1]
  const float* gam     = (const float*)d_in[5];   // scalar
  float* out = (float*)d_out;
  float* ws  = (float*)d_ws;

  const size_t SM4 = 4 * BUF * sizeof(float);   // ~66.6 KB
  const size_t SM5 = 5 * BUF * sizeof(float);   // ~83.2 KB

  k_init<<<1, 256, 0, stream>>>(ws);
  k_mean<<<256, 256, 0, stream>>>(X, ws + WS_G);
  k_sqrt_invsqrt<<<1, 256, SM4, stream>>>(ws + WS_G, ws + WS_GS, ws + WS_GIS);
  k_karcher_log<<<BATCH, 256, SM4, stream>>>(X, ws + WS_GIS, ws + WS_TACC);
  k_middle<<<1, 256, SM5, stream>>>(ws + WS_TACC, ws + WS_GS, rmean, gam, ws + WS_M);
  k_whiten_eigh<<<BATCH, 256, SM4, stream>>>(X, ws + WS_M, out, ws + WS_W, ws + WS_VAR);
  k_scalars<<<1, 1, 0, stream>>>(ws + WS_VAR, rvar, raw_std, gam, ws + WS_ALPHA);
  k_reconstruct<<<BATCH, 256, SM4, stream>>>(out, ws + WS_W, ws + WS_ALPHA, Q, out);
}